// LSTM_Layer_v1_12223476924617
// MI455X (gfx1250) — compile-verified
//
#include <hip/hip_runtime.h>

// LSTM layer for MI455X (gfx1250, wave32, WMMA).
// Strategy:
//   1) xW = x @ W + bias as one large parallel fp32 WMMA GEMM (M=131072, N=2048, K=512)
//      stored in workspace (falls back to fused per-step x@W if ws too small).
//   2) 2048 sequential step kernels. Each step splits K across 4 waves
//      (critical path: 128 dependent WMMAs instead of 512), reduces partials
//      through LDS, then applies sigmoid/tanh/c-update in registers.
//   3) tail copy of (h_T, c_T) into the output tail.

typedef __attribute__((ext_vector_type(2))) float v2f;
typedef __attribute__((ext_vector_type(8))) float v8f;

#define S_LEN 2048
#define BATCH 64
#define IN    512
#define HID   512
#define GATES 2048 /* 4*HID */

__device__ __forceinline__ float sigmoidf(float x) {
  return 1.0f / (1.0f + __expf(-x));
}

// One K-sweep of length Klen accumulating 4 output tiles that share the A matrix.
// A fragment (16x4 f32, 2 VGPRs): lane L, reg v -> (M = L%16, K = 2*(L/16)+v)
// B fragment (4x16 f32, 2 VGPRs): lane L, reg v -> (K = 2*(L/16)+v, N = L%16)
// Bmat is row-major [*, GATES]; the 4 tiles sit at columns coloff + g*tile_stride.
__device__ __forceinline__ void wmma_k_sweep(const float* __restrict__ arow,
                                             const float* __restrict__ Bmat,
                                             int Klen, int coloff, int tile_stride,
                                             int l16, int lh, v8f* acc) {
  for (int k = 0; k < Klen; k += 4) {
    v2f a = *(const v2f*)(arow + k);
    const float* bptr = Bmat + (size_t)(k + 2 * lh) * GATES + coloff + l16;
#pragma unroll
    for (int g = 0; g < 4; ++g) {
      v2f b;
      b.x = bptr[(size_t)g * tile_stride];
      b.y = bptr[(size_t)g * tile_stride + GATES];
      acc[g] = __builtin_amdgcn_wmma_f32_16x16x4_f32(
          false, a, false, b, (short)0, acc[g], false, false);
    }
  }
}

// ---------------------------------------------------------------------------
// Kernel 1: xw[S*B, GATES] = x[S*B, IN] @ W[IN, GATES] + bias
// block = 128 threads (4 waves); wave -> one 16-row M tile, 4 consecutive N tiles.
// grid = (8192/4, 2048/64) = (2048, 32)
// ---------------------------------------------------------------------------
__global__ void lstm_xw_gemm(const float* __restrict__ x,
                             const float* __restrict__ W,
                             const float* __restrict__ bias,
                             float* __restrict__ xw) {
  const int lane = threadIdx.x & 31;
  const int wave = threadIdx.x >> 5;
  const int l16  = lane & 15;
  const int lh   = lane >> 4;
  const int m0   = (blockIdx.x * 4 + wave) * 16;
  const int n0   = blockIdx.y * 64;

  v8f acc[4];
#pragma unroll
  for (int t = 0; t < 4; ++t) {
    float bv = bias[n0 + t * 16 + l16];
#pragma unroll
    for (int v = 0; v < 8; ++v) acc[t][v] = bv;
  }

  const float* arow = x + (size_t)(m0 + l16) * IN + 2 * lh;
  wmma_k_sweep(arow, W, IN, n0, 16, l16, lh, acc);

#pragma unroll
  for (int t = 0; t < 4; ++t)
#pragma unroll
    for (int v = 0; v < 8; ++v)
      xw[(size_t)(m0 + v + 8 * lh) * GATES + n0 + t * 16 + l16] = acc[t][v];
}

// ---------------------------------------------------------------------------
// Kernel 2: one recurrence step, K split across 4 waves to shorten the
// sequential critical path.
// block = 512 threads = 16 waves: wave w -> (mtile = w&3, kchunk = w>>2).
// grid = 32 (j tiles of 16 columns).
// gates[B, GATES] = (xw_t ? xw_t : bias + x_t@W) + h_prev@U, then elementwise.
// kchunk 0 seeds acc with xw_t/bias; kchunks 1..3 accumulate partials from
// their K range, dump them to LDS, and kchunk 0 reduces + applies gates.
// ---------------------------------------------------------------------------
__global__ void lstm_step(const float* __restrict__ xw_t,   // [B, GATES] or null
                          const float* __restrict__ x_t,    // [B, IN]
                          const float* __restrict__ W,      // [IN, GATES]
                          const float* __restrict__ h_prev, // [B, HID] or null (t==0)
                          const float* __restrict__ U,      // [HID, GATES]
                          const float* __restrict__ bias,   // [GATES]
                          float* __restrict__ c,            // [B, HID] in/out
                          float* __restrict__ h_out) {      // [B, HID]
  // Partial sums: [kchunk-1][mtile][gate][lane*8 + v]  (12 * 4 * 256 floats = 48 KB)
  __shared__ float red[12 * 4 * 256];

  const int lane  = threadIdx.x & 31;
  const int w     = threadIdx.x >> 5;
  const int mtile = w & 3;
  const int kc    = w >> 2;            // 0..3
  const int l16   = lane & 15;
  const int lh    = lane >> 4;
  const int m0    = mtile * 16;
  const int j0    = blockIdx.x * 16;

  v8f acc[4];  // gate tiles: i, f, g, o at columns j0 + {0,1,2,3}*HID
  if (kc == 0) {
    if (xw_t) {
#pragma unroll
      for (int g = 0; g < 4; ++g)
#pragma unroll
        for (int v = 0; v < 8; ++v)
          acc[g][v] = xw_t[(size_t)(m0 + v + 8 * lh) * GATES + g * HID + j0 + l16];
    } else {
#pragma unroll
      for (int g = 0; g < 4; ++g) {
        float bv = bias[g * HID + j0 + l16];
#pragma unroll
        for (int v = 0; v < 8; ++v) acc[g][v] = bv;
      }
    }
  } else {
#pragma unroll
    for (int g = 0; g < 4; ++g)
#pragma unroll
      for (int v = 0; v < 8; ++v) acc[g][v] = 0.0f;
  }

  if (xw_t) {
    // Recurrent product only: K=512 split into 4 chunks of 128.
    if (h_prev) {
      const int k0 = kc * 128;
      const float* arow = h_prev + (size_t)(m0 + l16) * HID + k0 + 2 * lh;
      wmma_k_sweep(arow, U + (size_t)k0 * GATES, 128, j0, HID, l16, lh, acc);
    }
  } else {
    // Fused fallback: total K=1024 (x@W then h@U), 256 per chunk.
    if (kc < 2) {
      const int k0 = kc * 256;
      const float* arow = x_t + (size_t)(m0 + l16) * IN + k0 + 2 * lh;
      wmma_k_sweep(arow, W + (size_t)k0 * GATES, 256, j0, HID, l16, lh, acc);
    } else if (h_prev) {
      const int k0 = (kc - 2) * 256;
      const float* arow = h_prev + (size_t)(m0 + l16) * HID + k0 + 2 * lh;
      wmma_k_sweep(arow, U + (size_t)k0 * GATES, 256, j0, HID, l16, lh, acc);
    }
  }

  if (kc > 0) {
    float* base = &red[((kc - 1) * 4 + mtile) * 4 * 256];
#pragma unroll
    for (int g = 0; g < 4; ++g)
      *(v8f*)&base[g * 256 + lane * 8] = acc[g];
  }
  __syncthreads();

  if (kc == 0) {
#pragma unroll
    for (int p = 0; p < 3; ++p) {
      const float* base = &red[(p * 4 + mtile) * 4 * 256];
#pragma unroll
      for (int g = 0; g < 4; ++g)
        acc[g] += *(const v8f*)&base[g * 256 + lane * 8];
    }
#pragma unroll
    for (int v = 0; v < 8; ++v) {
      const int m = m0 + v + 8 * lh;
      const int j = j0 + l16;
      const size_t idx = (size_t)m * HID + j;
      float ig = sigmoidf(acc[0][v]);
      float fg = sigmoidf(acc[1][v]);
      float gg = tanhf(acc[2][v]);
      float og = sigmoidf(acc[3][v]);
      float cn = fg * c[idx] + ig * gg;
      c[idx] = cn;
      h_out[idx] = og * tanhf(cn);
    }
  }
}

// ---------------------------------------------------------------------------
// Helpers: zero the cell state; copy (h_T, c_T) to the output tail.
// ---------------------------------------------------------------------------
__global__ void lstm_zero(float* __restrict__ p, int n) {
  int i = blockIdx.x * blockDim.x + threadIdx.x;
  if (i < n) p[i] = 0.0f;
}

__global__ void lstm_tail(const float* __restrict__ hT,
                          const float* __restrict__ cT,
                          float* __restrict__ tail) {
  int i = blockIdx.x * blockDim.x + threadIdx.x;
  if (i < BATCH * HID) {
    tail[i] = hT[i];
    tail[BATCH * HID + i] = cT[i];
  }
}

extern "C" void kernel_launch(void* const* d_in, const int* in_sizes, int n_in,
                              void* d_out, int out_size, void* d_ws, size_t ws_size,
                              hipStream_t stream) {
  (void)in_sizes; (void)n_in; (void)out_size;

  const float* x    = (const float*)d_in[0];  // [S, B, IN]
  const float* W    = (const float*)d_in[1];  // [IN, GATES]
  const float* U    = (const float*)d_in[2];  // [HID, GATES]
  const float* bias = (const float*)d_in[3];  // [GATES]
  float* out = (float*)d_out;                 // [S*B*HID] ++ [B*HID] ++ [B*HID]

  float* c_state = (float*)d_ws;                                   // [B, HID]
  float* xw = (float*)((char*)d_ws + (size_t)BATCH * HID * sizeof(float));
  const size_t need = (size_t)BATCH * HID * sizeof(float) +
                      (size_t)S_LEN * BATCH * GATES * sizeof(float);
  const bool precompute = (ws_size >= need);

  // Zero cell state (ws is poisoned by the harness).
  lstm_zero<<<(BATCH * HID + 255) / 256, 256, 0, stream>>>(c_state, BATCH * HID);

  if (precompute) {
    // Parallel half: xW for all timesteps in one WMMA GEMM.
    lstm_xw_gemm<<<dim3(2048, 32), 128, 0, stream>>>(x, W, bias, xw);
  }

  // Sequential recurrence: 2048 dependent step kernels (graph-captured).
  for (int t = 0; t < S_LEN; ++t) {
    const float* xw_t = precompute ? (xw + (size_t)t * BATCH * GATES) : nullptr;
    const float* x_t  = x + (size_t)t * BATCH * IN;
    const float* h_p  = (t > 0) ? (out + (size_t)(t - 1) * BATCH * HID) : nullptr;
    float* h_o        = out + (size_t)t * BATCH * HID;
    lstm_step<<<32, 512, 0, stream>>>(xw_t, x_t, W, h_p, U, bias, c_state, h_o);
  }

  // (h_T, c_T) tuple outputs appended after hidden_seq.
  lstm_tail<<<(BATCH * HID + 255) / 256, 256, 0, stream>>>(
      out + (size_t)(S_LEN - 1) * BATCH * HID, c_state,
      out + (size_t)S_LEN * BATCH * HID);
}